// HeuristicPositiveSampleAssigner_v2_33045478375658
// MI455X (gfx1250) — compile-verified
//
#include <hip/hip_runtime.h>
#include <math.h>

#define NC       80
#define TOPKN    13
#define NA_MAX   8400
#define NMAX_MAX 64
#define TILE     525          // 8400 = 16 * 525; tile = 525*16B = 8.4KB per buffer
#define EPS_A    1e-9f
#define CIOU_EPS 1e-7f
#define DYN_T_F  0.4f

typedef unsigned int u32x4 __attribute__((ext_vector_type(4)));
typedef int          i32x8 __attribute__((ext_vector_type(8)));
typedef int          i32x4 __attribute__((ext_vector_type(4)));

#if defined(__HIP_DEVICE_COMPILE__) && defined(__gfx1250__) && __has_builtin(__builtin_amdgcn_tensor_load_to_lds)
#define HAVE_TDM 1
#else
#define HAVE_TDM 0
#endif

__device__ __forceinline__ float ciou_gt_pd(float g0, float g1, float g2, float g3,
                                            float p0, float p1, float p2, float p3) {
    float w1 = g2 - g0, h1 = g3 - g1;
    float w2 = p2 - p0, h2 = p3 - p1;
    float iw = fminf(g2, p2) - fmaxf(g0, p0); iw = iw > 0.f ? iw : 0.f;
    float ih = fminf(g3, p3) - fmaxf(g1, p1); ih = ih > 0.f ? ih : 0.f;
    float inter = iw * ih;
    float uni = w1 * h1 + w2 * h2 - inter + CIOU_EPS;
    float iou = inter / uni;
    float cw = fmaxf(g2, p2) - fminf(g0, p0);
    float ch = fmaxf(g3, p3) - fminf(g1, p1);
    float c2 = cw * cw + ch * ch + CIOU_EPS;
    float dx = p0 + p2 - g0 - g2, dy = p1 + p3 - g1 - g3;
    float rho2 = (dx * dx + dy * dy) * 0.25f;
    const float FOUR_OVER_PI2 = 4.0f / (3.14159265358979323846f * 3.14159265358979323846f);
    float dat = atanf(w2 / h2) - atanf(w1 / h1);
    float v = FOUR_OVER_PI2 * dat * dat;
    float a = v / (v - iou + (1.0f + CIOU_EPS));
    return iou - (rho2 / c2 + v * a);
}

__device__ __forceinline__ float pow6(float o) {
    float o2 = o * o; float o4 = o2 * o2; return o4 * o2;
}

#if HAVE_TDM
// TDM: 1D contiguous copy of nelem 4-byte elements global->LDS
// (D# per CDNA5 ISA; 6-arg clang-23 builtin form)
__device__ __forceinline__ void tdm_load_dw(const void* gptr, unsigned ldsOff, int nelem) {
    unsigned long long ga = (unsigned long long)(size_t)gptr;
    u32x4 g0;
    g0[0] = 1u;                                    // count=1, user descriptor
    g0[1] = ldsOff;                                // lds_addr (bytes)
    g0[2] = (unsigned)(ga & 0xffffffffull);        // global_addr[31:0]
    g0[3] = (unsigned)(ga >> 32) | (2u << 30);     // global_addr[56:32] | type=2 (image)
    i32x8 g1;
    g1[0] = (2 << 16);                             // wg_mask=0, data_size=2 (4B)
    g1[1] = (nelem & 0xffff) << 16;                // tensor_dim0[15:0]
    g1[2] = ((nelem >> 16) & 0xffff) | (1 << 16);  // tensor_dim0[31:16] | tensor_dim1=1
    g1[3] = (nelem & 0xffff) << 16;                // tile_dim0 = nelem
    g1[4] = 1;                                     // tile_dim1=1, tile_dim2=0
    g1[5] = nelem;                                 // tensor_dim0_stride[31:0]
    g1[6] = 0;
    g1[7] = 0;
    i32x4 z4 = {0, 0, 0, 0};                       // groups 2/3 unused (<=2D tile)
    i32x8 z8 = {0, 0, 0, 0, 0, 0, 0, 0};
    __builtin_amdgcn_tensor_load_to_lds(g0, g1, z4, z4, z8, 0);
}
#endif

// ---------------------------------------------------------------------------
// Pass A: one block per (b, gt). Build align row (8400) in LDS with TDM-staged
// pd_bboxes tiles (double-buffered, overlapped with CIoU math), then select
// top-13 anchor indices (lowest-index tiebreak = jax.lax.top_k semantics).
// ---------------------------------------------------------------------------
__global__ __launch_bounds__(256) void assign_pass_a(
    const float* __restrict__ pd_scores, const float* __restrict__ pd_bboxes,
    const float* __restrict__ anc, const int* __restrict__ gt_labels,
    const float* __restrict__ gt_bboxes, const float* __restrict__ mask_gt,
    int* __restrict__ topk_out, unsigned* __restrict__ amax,
    int na, int nmax)
{
    __shared__ float sAlign[NA_MAX];      // 33.6 KB
    __shared__ float sBox[2 * TILE * 4];  // 16.8 KB double buffer
    __shared__ float sV[256];
    __shared__ int   sI[256];
    __shared__ float sGt[4];
    __shared__ int   sLab;
    __shared__ float sMgt;

    const int b = blockIdx.y, i = blockIdx.x, tid = threadIdx.x;
    const int gi = b * nmax + i;
    if (tid == 0) {
        sGt[0] = gt_bboxes[(size_t)gi * 4 + 0];
        sGt[1] = gt_bboxes[(size_t)gi * 4 + 1];
        sGt[2] = gt_bboxes[(size_t)gi * 4 + 2];
        sGt[3] = gt_bboxes[(size_t)gi * 4 + 3];
        sLab = gt_labels[gi];
        sMgt = mask_gt[gi];
        amax[gi] = 0u;                    // init per-gt align max for pass B/C
    }
    __syncthreads();
    const float gx0 = sGt[0], gy0 = sGt[1], gx1 = sGt[2], gy1 = sGt[3];
    const int lab = sLab;
    const float mgt = sMgt;
    const float* __restrict__ pb = pd_bboxes + (size_t)b * na * 4;
    const float* __restrict__ ps = pd_scores + (size_t)b * na * NC + lab;
    const int ntiles = (na + TILE - 1) / TILE;

#if HAVE_TDM
    if (tid < 32) {   // wave 0 issues TDM (EXEC-independent, one op per wave)
        int c0 = min(TILE, na);
        tdm_load_dw(pb, (unsigned)(size_t)(&sBox[0]), c0 * 4);
    }
#endif
    for (int t = 0; t < ntiles; ++t) {
        const int j0 = t * TILE;
        const int cnt = min(TILE, na - j0);
        float* buf = sBox + (t & 1) * TILE * 4;
#if HAVE_TDM
        if (tid < 32) __builtin_amdgcn_s_wait_tensorcnt(0);
        __syncthreads();                  // tile t visible to all waves
        if (tid < 32 && (t + 1) < ntiles) {
            int j0n = (t + 1) * TILE;
            int cn = min(TILE, na - j0n);
            tdm_load_dw(pb + (size_t)j0n * 4,
                        (unsigned)(size_t)(&sBox[((t + 1) & 1) * TILE * 4]), cn * 4);
        }
#else
        for (int k = tid; k < cnt * 4; k += 256) buf[k] = pb[(size_t)j0 * 4 + k];
        __syncthreads();
#endif
        for (int j = j0 + tid; j < j0 + cnt; j += 256) {
            float ax = anc[(size_t)j * 2 + 0];
            float ay = anc[(size_t)j * 2 + 1];
            int l = (j - j0) * 4;
            float p0 = buf[l], p1 = buf[l + 1], p2 = buf[l + 2], p3 = buf[l + 3];
            float dmin = fminf(fminf(ax - gx0, ay - gy0), fminf(gx1 - ax, gy1 - ay));
            float al = 0.f;
            if ((dmin > EPS_A) && (mgt > 0.f)) {
                float ci = ciou_gt_pd(gx0, gy0, gx1, gy1, p0, p1, p2, p3);
                float ov = ci > 0.f ? ci : 0.f;
                float sc = ps[(size_t)j * NC];
                al = sc * pow6(ov);
            }
            sAlign[j] = al;
        }
        __syncthreads();                  // also fences buffer reuse
    }

    // top-13 by iterative block argmax; ties -> lowest index (jax.lax.top_k)
    for (int k = 0; k < TOPKN; ++k) {
        float best = -1.0f; int bi = na;
        for (int j = tid; j < na; j += 256) {
            float v = sAlign[j];
            if (v > best) { best = v; bi = j; }
        }
        sV[tid] = best; sI[tid] = bi;
        __syncthreads();
        for (int s = 128; s > 0; s >>= 1) {
            if (tid < s) {
                float v2 = sV[tid + s]; int i2 = sI[tid + s];
                if (v2 > sV[tid] || (v2 == sV[tid] && i2 < sI[tid])) { sV[tid] = v2; sI[tid] = i2; }
            }
            __syncthreads();
        }
        if (tid == 0) {
            int w = sI[0];
            topk_out[(size_t)gi * TOPKN + k] = w;
            sAlign[w] = -2.0f;            // remove winner; all live values >= 0
        }
        __syncthreads();
    }
}

// ---------------------------------------------------------------------------
// Pass B: one thread per (b, anchor). GT tables staged to LDS via TDM (4
// chained descriptors, one s_wait_tensorcnt). Resolve assignment across gts,
// write labels/bboxes/fg/tgt_idx outputs, accumulate per-gt align max.
// ---------------------------------------------------------------------------
__global__ __launch_bounds__(256) void assign_pass_b(
    const float* __restrict__ pd_scores, const float* __restrict__ pd_bboxes,
    const float* __restrict__ anc, const int* __restrict__ gt_labels,
    const float* __restrict__ gt_bboxes, const float* __restrict__ mask_gt,
    const int* __restrict__ topk_in, unsigned* __restrict__ amax,
    float* __restrict__ out_labels, float* __restrict__ out_bboxes,
    float* __restrict__ out_fg, float* __restrict__ out_tgt,
    float* __restrict__ alignval, int* __restrict__ labsel, int* __restrict__ gsel,
    int na, int nmax)
{
    __shared__ float sB[NMAX_MAX * 4];
    __shared__ int   sL[NMAX_MAX];
    __shared__ float sM[NMAX_MAX];
    __shared__ int   sT[NMAX_MAX * TOPKN];

    const int b = blockIdx.y, tid = threadIdx.x;
#if HAVE_TDM
    if (tid < 32) {   // wave 0: 4 TDM descriptors, in-order on TENSORcnt
        tdm_load_dw(gt_bboxes + (size_t)b * nmax * 4, (unsigned)(size_t)(&sB[0]), nmax * 4);
        tdm_load_dw(gt_labels + (size_t)b * nmax,     (unsigned)(size_t)(&sL[0]), nmax);
        tdm_load_dw(mask_gt   + (size_t)b * nmax,     (unsigned)(size_t)(&sM[0]), nmax);
        tdm_load_dw(topk_in   + (size_t)b * nmax * TOPKN, (unsigned)(size_t)(&sT[0]), nmax * TOPKN);
        __builtin_amdgcn_s_wait_tensorcnt(0);
    }
#else
    for (int k = tid; k < nmax * 4; k += 256) sB[k] = gt_bboxes[(size_t)b * nmax * 4 + k];
    for (int k = tid; k < nmax; k += 256) { sL[k] = gt_labels[b * nmax + k]; sM[k] = mask_gt[b * nmax + k]; }
    for (int k = tid; k < nmax * TOPKN; k += 256) sT[k] = topk_in[(size_t)b * nmax * TOPKN + k];
#endif
    __syncthreads();

    const int j = blockIdx.x * 256 + tid;
    if (j >= na) return;

    const float ax = anc[(size_t)j * 2 + 0];
    const float ay = anc[(size_t)j * 2 + 1];
    const float* pbj = pd_bboxes + ((size_t)b * na + j) * 4;
    const float p0 = pbj[0], p1 = pbj[1], p2 = pbj[2], p3 = pbj[3];

    int fg = 0, gSingle = 0, bestI = 0;
    float bestOv = 0.f, ovSingle = 0.f;
    for (int i = 0; i < nmax; ++i) {
        float gx0 = sB[i * 4 + 0], gy0 = sB[i * 4 + 1], gx1 = sB[i * 4 + 2], gy1 = sB[i * 4 + 3];
        float dmin = fminf(fminf(ax - gx0, ay - gy0), fminf(gx1 - ax, gy1 - ay));
        bool m = (dmin > EPS_A) && (sM[i] > 0.f);
        float ov = 0.f;
        if (m) {
            float ci = ciou_gt_pd(gx0, gy0, gx1, gy1, p0, p1, p2, p3);
            ov = ci > 0.f ? ci : 0.f;
        }
        if (i == 0) { bestOv = ov; bestI = 0; }
        else if (ov > bestOv) { bestOv = ov; bestI = i; }   // first-max = jnp.argmax
        bool member = false;
        #pragma unroll
        for (int t2 = 0; t2 < TOPKN; ++t2) member = member || (sT[i * TOPKN + t2] == j);
        if (member && m) {
            if (fg == 0) { gSingle = i; ovSingle = ov; }
            ++fg;
        }
    }

    int g; float fgF, ovg;
    if (fg > 1)      { g = bestI;   fgF = 1.f; ovg = bestOv;   }
    else if (fg == 1){ g = gSingle; fgF = 1.f; ovg = ovSingle; }
    else             { g = 0;       fgF = 0.f; ovg = 0.f;      }

    const int lg = sL[g];
    const size_t bj = (size_t)b * na + j;
    out_labels[bj] = (float)lg;
    out_bboxes[bj * 4 + 0] = sB[g * 4 + 0];
    out_bboxes[bj * 4 + 1] = sB[g * 4 + 1];
    out_bboxes[bj * 4 + 2] = sB[g * 4 + 2];
    out_bboxes[bj * 4 + 3] = sB[g * 4 + 3];
    out_fg[bj]  = fgF;
    out_tgt[bj] = (float)g;

    float al = 0.f;
    if (fgF > 0.f) {
        float sc = pd_scores[bj * NC + lg];
        al = sc * pow6(ovg);
        atomicMax(&amax[b * nmax + g], __float_as_uint(al));  // al >= 0 -> exact
    }
    alignval[bj] = al;
    labsel[bj]   = (fgF > 0.f) ? lg : -1;
    gsel[bj]     = g;
}

// ---------------------------------------------------------------------------
// Pass C: one thread per target_scores element (86 MB streaming write).
// ---------------------------------------------------------------------------
__global__ __launch_bounds__(256) void assign_pass_c(
    const float* __restrict__ alignval, const int* __restrict__ labsel,
    const int* __restrict__ gsel, const unsigned* __restrict__ amax,
    float* __restrict__ out_scores, int na, int nmax, long long total)
{
    long long e = (long long)blockIdx.x * 256 + threadIdx.x;
    if (e >= total) return;
    long long bj = e / NC;
    int c = (int)(e - bj * NC);
    float out = 0.f;
    int lab = labsel[bj];
    if (lab == c) {
        int b = (int)(bj / na);
        int g = gsel[bj];
        float am = __uint_as_float(amax[b * nmax + g]);
        out = alignval[bj] / (DYN_T_F * am + EPS_A);
    }
    out_scores[e] = out;
}

extern "C" void kernel_launch(void* const* d_in, const int* in_sizes, int n_in,
                              void* d_out, int out_size, void* d_ws, size_t ws_size,
                              hipStream_t stream) {
    const float* pd_scores = (const float*)d_in[0];
    const float* pd_bboxes = (const float*)d_in[1];
    const float* anc       = (const float*)d_in[2];
    const int*   gt_labels = (const int*)d_in[3];
    const float* gt_bboxes = (const float*)d_in[4];
    const float* mask_gt   = (const float*)d_in[5];

    const int na   = in_sizes[2] / 2;           // anc_points (na,2)
    const int bs   = in_sizes[1] / (na * 4);    // pd_bboxes (bs,na,4)
    const int nmax = in_sizes[3] / bs;          // gt_labels (bs,nmax,1)
    const size_t bn  = (size_t)bs * nmax;
    const size_t bna = (size_t)bs * na;

    // workspace layout (~3.3 MB)
    char* w = (char*)d_ws;
    size_t off = 0;
    auto carve = [&](size_t bytes) { size_t o = off; off += (bytes + 255) & ~(size_t)255; return o; };
    int*      topk     = (int*)(w + carve(bn * TOPKN * sizeof(int)));
    unsigned* amax     = (unsigned*)(w + carve(bn * sizeof(unsigned)));
    float*    alignval = (float*)(w + carve(bna * sizeof(float)));
    int*      labsel   = (int*)(w + carve(bna * sizeof(int)));
    int*      gsel     = (int*)(w + carve(bna * sizeof(int)));

    // output layout: labels | bboxes | scores | fg | tgt_idx (flat, floats)
    float* out = (float*)d_out;
    float* oL = out;
    float* oB = oL + bna;
    float* oS = oB + bna * 4;
    float* oF = oS + bna * NC;
    float* oT = oF + bna;

    dim3 gA((unsigned)nmax, (unsigned)bs);
    assign_pass_a<<<gA, 256, 0, stream>>>(pd_scores, pd_bboxes, anc, gt_labels,
                                          gt_bboxes, mask_gt, topk, amax, na, nmax);

    dim3 gB((unsigned)((na + 255) / 256), (unsigned)bs);
    assign_pass_b<<<gB, 256, 0, stream>>>(pd_scores, pd_bboxes, anc, gt_labels,
                                          gt_bboxes, mask_gt, topk, amax,
                                          oL, oB, oF, oT, alignval, labsel, gsel, na, nmax);

    long long total = (long long)bna * NC;
    dim3 gC((unsigned)((total + 255) / 256));
    assign_pass_c<<<gC, 256, 0, stream>>>(alignval, labsel, gsel, amax, oS, na, nmax, total);
}